// SPDPower_13546326851718
// MI455X (gfx1250) — compile-verified
//
#include <hip/hip_runtime.h>
#include <hip/hip_bf16.h>

// ---------------------------------------------------------------------------
// SPD matrix power  out[b] = U diag(s^w) U^T  where  X[b] = U diag(s) U^T.
//
// Spectral Chebyshev expansion of f(t)=t^w evaluated on the matrix itself
// (Clenshaw recurrence): every step is a batched 256^3 FP32 GEMM on
// V_WMMA_F32_16X16X4_F32.  Panel staging global->LDS is done by the Tensor
// Data Mover (tensor_load_to_lds, TENSORcnt, s_wait_tensorcnt), double
// buffered, with TDM hardware padding providing the bank-conflict-free LDS
// row pitch.  weight[0] is used as the (scalar) exponent; for weight==1 the
// quadrature coefficients are exact (c0,c1 only) and the result equals X.
// ---------------------------------------------------------------------------

typedef __attribute__((ext_vector_type(2))) float v2f;
typedef __attribute__((ext_vector_type(8))) float v8f;
typedef __attribute__((ext_vector_type(4))) unsigned int v4u;
typedef __attribute__((ext_vector_type(8))) int v8i_;
typedef __attribute__((ext_vector_type(4))) int v4i_;

namespace {
constexpr int kN      = 256;  // matrix dimension
constexpr int kBatch  = 512;  // batch count
constexpr int kTerms  = 16;   // Chebyshev terms c_0..c_15
constexpr int kQuad   = 64;   // Chebyshev-Gauss quadrature nodes
constexpr int kTileM  = 128;  // workgroup C-tile rows
constexpr int kTileN  = 64;   // workgroup C-tile cols
constexpr int kKC     = 32;   // K chunk staged through LDS (pow2 for TDM pad)
constexpr float kPi   = 3.14159265358979323846f;
constexpr size_t kParamBytes = 65536; // lo[B], hi[B], cheb[B*kTerms] (padded)
}

// --- TDM: issue a 2D tile load global->LDS (one call per wave; EXEC-ignored,
// issued once; tracked with TENSORcnt).  Descriptor per CDNA5 ISA ch.8. ---
__device__ __forceinline__ void tdm_load_2d(const void* gptr, unsigned lds_addr,
                                            unsigned tile_x, unsigned tile_y,
                                            unsigned long long row_stride_elts,
                                            unsigned pad_interval_code,
                                            unsigned pad_amount_code) {
  const unsigned long long ga = (unsigned long long)(uintptr_t)gptr;
  v4u g0;
  g0[0] = 1u;                                    // count=1, user descriptor
  g0[1] = lds_addr;                              // D#.lds_addr (bytes)
  g0[2] = (unsigned)(ga & 0xFFFFFFFFu);          // global_addr[31:0]
  g0[3] = (unsigned)((ga >> 32) & 0x01FFFFFFu)   // global_addr[56:32]
          | (2u << 30);                          // type = 2 ("image")
  v8i_ g1;
  g1[0] = (int)((2u << 16)                       // data_size = 4 bytes
                | (1u << 20)                     // pad_enable (load only)
                | (pad_interval_code << 22)      // pad every 2^(c+1) DWORDs
                | (pad_amount_code << 25));      // pad (c+1) DWORDs
  // tensor dims set large (no OOB clipping ever occurs; tiles are interior)
  const unsigned tdim = 0x40000u;                // 256K elements
  g1[1] = (int)((tdim & 0xFFFFu) << 16);         // abar_addr=0, dim0 lo16
  g1[2] = (int)(((tdim >> 16) & 0xFFFFu) | ((tdim & 0xFFFFu) << 16));
  g1[3] = (int)(((tdim >> 16) & 0xFFFFu) | ((tile_x & 0xFFFFu) << 16));
  g1[4] = (int)(tile_y & 0xFFFFu);               // tile_dim1; tile_dim2 = 0
  g1[5] = (int)(unsigned)(row_stride_elts & 0xFFFFFFFFull);   // dim0_stride lo
  g1[6] = (int)(unsigned)((row_stride_elts >> 32) & 0xFFFFull); // hi16; d1s=0
  g1[7] = 0;
  const v4i_ z4 = {};
  const v8i_ z8 = {};
  // amdgpu-toolchain (clang-23) 6-arg form: (g0, g1, g2, g3, extra, cpol)
  __builtin_amdgcn_tensor_load_to_lds(g0, g1, z4, z4, z8, 0);
}

// --- Pass 1: per-batch spectral bounds (Gershgorin via symmetric column
// sums, coalesced) + per-batch Chebyshev coefficients of t^w on [lo,hi]. ---
__global__ __launch_bounds__(256) void spdpow_bounds_cheb(
    const float* __restrict__ X, const float* __restrict__ wgt,
    float* __restrict__ params) {
  const int b   = blockIdx.x;
  const int tid = threadIdx.x;
  const size_t base = (size_t)b * kN * kN;

  __shared__ float red[256];
  __shared__ float fv[kQuad];

  float s = 0.0f;
  for (int j = 0; j < kN; ++j) s += fabsf(X[base + (size_t)j * kN + tid]);
  red[tid] = s;
  __syncthreads();
  for (int off = 128; off > 0; off >>= 1) {
    if (tid < off) red[tid] = fmaxf(red[tid], red[tid + off]);
    __syncthreads();
  }
  const float hi = red[0] * 1.0001f + 1e-30f;  // upper eigenvalue bound
  const float lo = hi * 1e-6f;                 // crude positive lower bound
  const float w  = wgt[0];

  if (tid < kQuad) {
    const float th = kPi * ((float)tid + 0.5f) / (float)kQuad;
    const float t  = __cosf(th);
    const float xv = 0.5f * (t + 1.0f) * (hi - lo) + lo;
    fv[tid] = __expf(w * __logf(xv));          // f = x^w at quadrature node
  }
  __syncthreads();
  if (tid < kTerms) {
    float c = 0.0f;
    for (int j = 0; j < kQuad; ++j) {
      const float th = kPi * ((float)j + 0.5f) / (float)kQuad;
      c += fv[j] * __cosf((float)tid * th);
    }
    params[2 * kBatch + b * kTerms + tid] = c * (2.0f / (float)kQuad);
  }
  if (tid == 0) { params[b] = lo; params[kBatch + b] = hi; }
}

// --- Init: b_Kmax = c_Kmax * I ---
__global__ __launch_bounds__(256) void spdpow_init_diag(
    float* __restrict__ Out, const float* __restrict__ params, int cIdx) {
  const size_t i = (size_t)blockIdx.x * 256 + threadIdx.x;
  const int b  = (int)(i >> 16);
  const int rc = (int)(i & 0xFFFF);
  const float cv = params[2 * kBatch + b * kTerms + cIdx];
  Out[i] = ((rc >> 8) == (rc & 0xFF)) ? cv : 0.0f;
}

// --- Clenshaw GEMM step:
//   Out = mulT*(alpha*X@Bcur + beta*Bcur) + mulPrev*Bprev + cScale*c_k*I
// One workgroup = 128x64 C tile of one batch matrix; 8 wave32s in a
// 4(M) x 2(N) grid, each owning 2x2 WMMA 16x16 tiles.  Panels staged by the
// Tensor Data Mover with double buffering. ---
__global__ __launch_bounds__(256, 2) void spdpow_cheb_gemm(
    const float* __restrict__ X, const float* __restrict__ Bcur,
    const float* __restrict__ Bprev, float* __restrict__ Out,
    const float* __restrict__ params, float mulT, float mulPrev,
    int cIdx, float cScale) {
  const int b     = blockIdx.z;
  const int tileM = blockIdx.y * kTileM;
  const int tileN = blockIdx.x * kTileN;
  const size_t base = (size_t)b * kN * kN;
  const int tid   = threadIdx.x;
  const int lane  = tid & 31;
  const int wave  = tid >> 5;
  const int waveM = wave >> 1;  // 0..3 : 32-row band
  const int waveN = wave & 1;   // 0..1 : 32-col band

  // TDM pad codes: X rows 32 DWORDs + 1 pad  (interval 2^(4+1)=32, amount 1)
  //                B rows 64 DWORDs + 1 pad  (interval 2^(5+1)=64, amount 1)
  __shared__ float Xs[2][kTileM][kKC + 1];
  __shared__ float Bs[2][kKC][kTileN + 1];

  // The TDM writes LDS behind the compiler's back; plant opaque (never
  // executed) volatile stores so the ds_loads below are not folded to undef.
  int neverTrue;
  asm volatile("s_mov_b32 %0, 0" : "=s"(neverTrue));
  if (neverTrue) {
    ((volatile float*)Xs)[tid] = 1.0f;
    ((volatile float*)Bs)[tid] = 1.0f;
  }

  if (mulPrev != 0.0f)  // warm the epilogue operand into cache early
    __builtin_prefetch(&Bprev[base + (size_t)(tileM + (tid >> 1)) * kN + tileN], 0, 1);

  v8f acc[2][2] = {};

  // Issue chunk 0 into buffer 0 (one wave only; TDM issues once per wave).
  if (wave == 0) {
    tdm_load_2d(&X[base + (size_t)tileM * kN + 0],
                (unsigned)(uintptr_t)&Xs[0][0][0], kKC, kTileM, kN, 4u, 0u);
    tdm_load_2d(&Bcur[base + (size_t)0 * kN + tileN],
                (unsigned)(uintptr_t)&Bs[0][0][0], kTileN, kKC, kN, 5u, 0u);
  }

  for (int kc = 0, buf = 0; kc < kN; kc += kKC, buf ^= 1) {
    const bool more = (kc + kKC) < kN;
    if (wave == 0) {
      if (more) {  // prefetch next chunk into the other buffer
        tdm_load_2d(&X[base + (size_t)tileM * kN + (kc + kKC)],
                    (unsigned)(uintptr_t)&Xs[buf ^ 1][0][0],
                    kKC, kTileM, kN, 4u, 0u);
        tdm_load_2d(&Bcur[base + (size_t)(kc + kKC) * kN + tileN],
                    (unsigned)(uintptr_t)&Bs[buf ^ 1][0][0],
                    kTileN, kKC, kN, 5u, 0u);
        __builtin_amdgcn_s_wait_tensorcnt(2);  // current chunk landed
      } else {
        __builtin_amdgcn_s_wait_tensorcnt(0);
      }
    }
    __syncthreads();  // publish TDM-written LDS to all waves

    // WMMA f32 16x16x4 lane layouts:
    //  A 16x4 : lane L (L<16): M=L, K={0,1}; (L>=16): M=L-16, K={2,3}
    //  B 4x16 : lane L (L<16): N=L, K={0,1}; (L>=16): N=L-16, K={2,3}
    const int mrow0 = waveM * 32 + (lane & 15);
    const int ncol0 = waveN * 32 + (lane & 15);
    const int khalf = (lane >> 4) * 2;

    #pragma unroll
    for (int kk = 0; kk < kKC; kk += 4) {
      v2f afrag[2];
      v2f bfrag[2];
      #pragma unroll
      for (int mt = 0; mt < 2; ++mt) {
        afrag[mt].x = Xs[buf][mrow0 + mt * 16][kk + khalf + 0];
        afrag[mt].y = Xs[buf][mrow0 + mt * 16][kk + khalf + 1];
      }
      #pragma unroll
      for (int nt = 0; nt < 2; ++nt) {
        bfrag[nt].x = Bs[buf][kk + khalf + 0][ncol0 + nt * 16];
        bfrag[nt].y = Bs[buf][kk + khalf + 1][ncol0 + nt * 16];
      }
      #pragma unroll
      for (int mt = 0; mt < 2; ++mt)
        #pragma unroll
        for (int nt = 0; nt < 2; ++nt)
          acc[mt][nt] = __builtin_amdgcn_wmma_f32_16x16x4_f32(
              false, afrag[mt], false, bfrag[nt],
              (short)0, acc[mt][nt], false, false);
    }
    __syncthreads();  // all waves done with buf before TDM refills it
  }

  // Fused Clenshaw epilogue.
  const float lo    = params[b];
  const float hi    = params[kBatch + b];
  const float alpha = 2.0f / (hi - lo);
  const float beta  = -(hi + lo) / (hi - lo);
  const float cdiag = cScale * params[2 * kBatch + b * kTerms + cIdx];
  const bool  usePrev = (mulPrev != 0.0f);

  #pragma unroll
  for (int mt = 0; mt < 2; ++mt) {
    #pragma unroll
    for (int nt = 0; nt < 2; ++nt) {
      #pragma unroll
      for (int r = 0; r < 8; ++r) {
        // C/D 16x16 layout: VGPR r, lanes<16: M=r, lanes>=16: M=r+8
        const int gm = tileM + waveM * 32 + mt * 16 + r + ((lane >> 4) << 3);
        const int gn = tileN + waveN * 32 + nt * 16 + (lane & 15);
        const size_t idx = base + (size_t)gm * kN + gn;
        float v = mulT * (alpha * acc[mt][nt][r] + beta * Bcur[idx]);
        if (usePrev) v += mulPrev * Bprev[idx];  // in-place safe: 1 RW/elem
        if (gm == gn) v += cdiag;
        Out[idx] = v;
      }
    }
  }
}

extern "C" void kernel_launch(void* const* d_in, const int* in_sizes, int n_in,
                              void* d_out, int out_size, void* d_ws, size_t ws_size,
                              hipStream_t stream) {
  (void)in_sizes; (void)n_in; (void)out_size; (void)ws_size;
  const float* X  = (const float*)d_in[0];
  const float* Wt = (const float*)d_in[1];

  float* params = (float*)d_ws;                          // 64 KB param block
  float* bufA   = (float*)((char*)d_ws + kParamBytes);   // 128 MB matrix buf
  float* bufO   = (float*)d_out;                         // 128 MB matrix buf

  spdpow_bounds_cheb<<<kBatch, 256, 0, stream>>>(X, Wt, params);

  // Clenshaw: b_k = 2*Xt*b_{k+1} - b_{k+2} + c_k*I, parity-assigned buffers
  // so b_even lives in d_out and the final value lands in d_out.
  auto tgt = [&](int k) -> float* { return (k & 1) ? bufA : bufO; };
  const int Kmax = kTerms - 1;

  const size_t total = (size_t)kBatch * kN * kN;
  spdpow_init_diag<<<(unsigned)(total / 256), 256, 0, stream>>>(
      tgt(Kmax), params, Kmax);

  const dim3 grid(kN / kTileN, kN / kTileM, kBatch);
  for (int k = Kmax - 1; k >= 1; --k) {
    const bool hasPrev = (k + 2 <= Kmax);
    spdpow_cheb_gemm<<<grid, 256, 0, stream>>>(
        X, tgt(k + 1), hasPrev ? tgt(k + 2) : tgt(k + 1), tgt(k), params,
        2.0f, hasPrev ? -1.0f : 0.0f, k, 1.0f);
  }
  // p(Xt) = Xt*b_1 - b_2 + 0.5*c_0*I  -> d_out (reads b_2 from d_out in place)
  spdpow_cheb_gemm<<<grid, 256, 0, stream>>>(
      X, tgt(1), tgt(2), bufO, params, 1.0f, -1.0f, 0, 0.5f);
}